// TransformerGAT_57973468562009
// MI455X (gfx1250) — compile-verified
//
#include <hip/hip_runtime.h>
#include <hip/hip_bf16.h>
#include <math.h>

#define N_NODES 100000
#define N_EDGES 1600000
#define F_IN    256
#define F_HID   128
#define F_LAT   2
#define N_SAMP  512
#define LEAKY   0.2f
#define ORD_NEG_INF 0x007FFFFFu  // monotonic encoding of -inf

typedef float vf2 __attribute__((ext_vector_type(2)));
typedef float vf8 __attribute__((ext_vector_type(8)));

// ---------- helpers ----------
__device__ __forceinline__ unsigned f2ord(float f) {
  unsigned u = __float_as_uint(f);
  return (u & 0x80000000u) ? ~u : (u | 0x80000000u);
}
__device__ __forceinline__ float ord2f(unsigned u) {
  unsigned b = (u & 0x80000000u) ? (u & 0x7FFFFFFFu) : ~u;
  return __uint_as_float(b);
}
__device__ __forceinline__ float leaky_relu(float v) { return v > 0.f ? v : LEAKY * v; }
__device__ __forceinline__ float wave_reduce(float v) {
#pragma unroll
  for (int off = 16; off > 0; off >>= 1) v += __shfl_xor(v, off, 32);
  return v;
}
__device__ __forceinline__ void wait_asynccnt0() {
#if __has_builtin(__builtin_amdgcn_s_wait_asynccnt)
  __builtin_amdgcn_s_wait_asynccnt(0);
#else
  asm volatile("s_wait_asynccnt 0x0" ::: "memory");
#endif
}

// ---------- fill kernels (graph-replay-safe re-init) ----------
__global__ void fill_f32(float* p, float v, int n) {
  int i = blockIdx.x * blockDim.x + threadIdx.x;
  if (i < n) p[i] = v;
}
__global__ void fill_u32(unsigned* p, unsigned v, int n) {
  int i = blockIdx.x * blockDim.x + threadIdx.x;
  if (i < n) p[i] = v;
}

// ---------- GEMM: H[N,128] = X[N,256] @ W1[256,128] via V_WMMA_F32_16X16X4_F32 ----------
// One block = one 16-row band; 8 waves cover the 8 col-tiles (128 cols).
// X band is staged global->LDS with GLOBAL_LOAD_ASYNC_TO_LDS_B128 (ASYNCcnt path).
__global__ __launch_bounds__(256) void gemm_x_w1(const float* __restrict__ X,
                                                 const float* __restrict__ W,
                                                 float* __restrict__ H) {
  __shared__ float Xs[16 * F_IN];  // 16 KB
  const int band = blockIdx.x * 16;
  const int tid  = threadIdx.x;

  // prefetch next band into cache hierarchy (global_prefetch_b8)
  if (blockIdx.x + 1 < gridDim.x)
    __builtin_prefetch(X + (size_t)(band + 16) * F_IN + tid * 16, 0, 0);

  // async-stage the 16x256 f32 X band: 4 rounds x 256 lanes x 16B = 16 KB.
  // LDS generic-pointer low 32 bits == LDS byte offset (ISA 10.2 aperture rule).
#pragma unroll
  for (int i = tid * 4; i < 16 * F_IN; i += 256 * 4) {
    unsigned loff = (unsigned)(size_t)(&Xs[i]);
    unsigned long long ga =
        (unsigned long long)(X + (size_t)band * F_IN + i);
    asm volatile("global_load_async_to_lds_b128 %0, %1, off"
                 :: "v"(loff), "v"(ga) : "memory");
  }
  wait_asynccnt0();   // my wave's async LDS writes are visible
  __syncthreads();    // all waves' stages complete

  const int wave = tid >> 5;
  const int lane = tid & 31;
  const int hi   = lane >> 4;   // 0: lanes 0-15, 1: lanes 16-31
  const int l15  = lane & 15;
  const int c0   = wave * 16;   // output col tile

  vf8 acc = {};
  for (int kk = 0; kk < F_IN; kk += 4) {
    // A 16x4 f32: lane = M (l15), VGPR0/1 = K pair {0,1} (lo half) / {2,3} (hi half)
    const int ka = kk + (hi << 1);
    vf2 a, b;
    a.x = Xs[l15 * F_IN + ka];
    a.y = Xs[l15 * F_IN + ka + 1];
    // B 4x16 f32: row striped across lanes per VGPR; hi half-wave = K+2
    b.x = W[(size_t)ka * F_HID + c0 + l15];
    b.y = W[(size_t)(ka + 1) * F_HID + c0 + l15];
    acc = __builtin_amdgcn_wmma_f32_16x16x4_f32(false, a, false, b, (short)0, acc,
                                                false, false);
  }
  // C/D 16x16 f32: VGPR v -> row v (+8 for hi half-wave), col = l15
#pragma unroll
  for (int v = 0; v < 8; ++v) {
    const int row = band + v + hi * 8;
    H[(size_t)row * F_HID + c0 + l15] = acc[v];
  }
}

// ---------- per-node attention scalars: es[n]=h.a_src, ed[n]=h.a_dst ----------
__global__ __launch_bounds__(256) void node_edge_feats(const float* __restrict__ H,
                                                       const float* __restrict__ a_s,
                                                       const float* __restrict__ a_d,
                                                       float* __restrict__ es,
                                                       float* __restrict__ ed) {
  const int wave = threadIdx.x >> 5, lane = threadIdx.x & 31;
  const int n = blockIdx.x * 8 + wave;
  if (n >= N_NODES) return;
  const float4 h  = ((const float4*)(H + (size_t)n * F_HID))[lane];
  const float4 s4 = ((const float4*)a_s)[lane];
  const float4 d4 = ((const float4*)a_d)[lane];
  float ss = h.x * s4.x + h.y * s4.y + h.z * s4.z + h.w * s4.w;
  float dd = h.x * d4.x + h.y * d4.y + h.z * d4.z + h.w * d4.w;
  ss = wave_reduce(ss);
  dd = wave_reduce(dd);
  if (lane == 0) { es[n] = ss; ed[n] = dd; }
}

// ---------- edge pass 1: segment max (monotonic-uint atomicMax) ----------
__global__ void edge_max(const int* __restrict__ src, const int* __restrict__ dst,
                         const float* __restrict__ es, const float* __restrict__ ed,
                         unsigned* __restrict__ emax) {
  int e = blockIdx.x * blockDim.x + threadIdx.x;
  if (e >= N_EDGES) return;
  const int s = src[e], d = dst[e];
  const float v = leaky_relu(es[s] + ed[d]);
  atomicMax(&emax[d], f2ord(v));
}

// ---------- decode max in place: -inf (no in-edges) -> 0 ----------
__global__ void finalize_max(unsigned* __restrict__ emax, int n_nodes) {
  int n = blockIdx.x * blockDim.x + threadIdx.x;
  if (n >= n_nodes) return;
  float f = ord2f(emax[n]);
  if (!isfinite(f)) f = 0.f;
  ((float*)emax)[n] = f;
}

// ---------- edge pass 2: denom[d] += exp(e - max[d]) ----------
__global__ void edge_expsum(const int* __restrict__ src, const int* __restrict__ dst,
                            const float* __restrict__ es, const float* __restrict__ ed,
                            const float* __restrict__ emaxf, float* __restrict__ denom) {
  int e = blockIdx.x * blockDim.x + threadIdx.x;
  if (e >= N_EDGES) return;
  const int s = src[e], d = dst[e];
  const float v = leaky_relu(es[s] + ed[d]);
  atomicAdd(&denom[d], expf(v - emaxf[d]));
}

// ---------- edge pass 3 (F=128): AGG[d] += alpha * H[s], one wave per edge ----------
__global__ __launch_bounds__(256) void edge_aggregate128(
    const int* __restrict__ src, const int* __restrict__ dst,
    const float* __restrict__ es, const float* __restrict__ ed,
    const float* __restrict__ emaxf, const float* __restrict__ denom,
    const float* __restrict__ H, float* __restrict__ AGG) {
  const int wave = threadIdx.x >> 5, lane = threadIdx.x & 31;
  const int e = blockIdx.x * 8 + wave;
  if (e >= N_EDGES) return;
  const int s = src[e], d = dst[e];
  const float v = leaky_relu(es[s] + ed[d]);
  const float alpha = expf(v - emaxf[d]) / fmaxf(denom[d], 1e-16f);
  const float4 h = ((const float4*)(H + (size_t)s * F_HID))[lane];
  float* o = AGG + (size_t)d * F_HID + lane * 4;
  atomicAdd(o + 0, alpha * h.x);
  atomicAdd(o + 1, alpha * h.y);
  atomicAdd(o + 2, alpha * h.z);
  atomicAdd(o + 3, alpha * h.w);
}

// ---------- layer 2 prep: h2 = relu(AGG) @ Wmu (128->2), e2 scalars ----------
__global__ __launch_bounds__(256) void node_h2(const float* __restrict__ AGG,
                                               const float* __restrict__ Wmu,
                                               const float* __restrict__ amu_s,
                                               const float* __restrict__ amu_d,
                                               float* __restrict__ H2,
                                               float* __restrict__ e2s,
                                               float* __restrict__ e2d) {
  const int wave = threadIdx.x >> 5, lane = threadIdx.x & 31;
  const int n = blockIdx.x * 8 + wave;
  if (n >= N_NODES) return;
  float4 h = ((const float4*)(AGG + (size_t)n * F_HID))[lane];
  h.x = fmaxf(h.x, 0.f); h.y = fmaxf(h.y, 0.f);
  h.z = fmaxf(h.z, 0.f); h.w = fmaxf(h.w, 0.f);
  const int k = lane * 4;
  float m0 = h.x * Wmu[(k + 0) * 2]     + h.y * Wmu[(k + 1) * 2]
           + h.z * Wmu[(k + 2) * 2]     + h.w * Wmu[(k + 3) * 2];
  float m1 = h.x * Wmu[(k + 0) * 2 + 1] + h.y * Wmu[(k + 1) * 2 + 1]
           + h.z * Wmu[(k + 2) * 2 + 1] + h.w * Wmu[(k + 3) * 2 + 1];
  m0 = wave_reduce(m0);
  m1 = wave_reduce(m1);
  if (lane == 0) {
    H2[n * 2 + 0] = m0;
    H2[n * 2 + 1] = m1;
    e2s[n] = m0 * amu_s[0] + m1 * amu_s[1];
    e2d[n] = m0 * amu_d[0] + m1 * amu_d[1];
  }
}

// ---------- edge pass 3 (F=2): MU[d] += alpha * H2[s] ----------
__global__ void edge_aggregate2(const int* __restrict__ src, const int* __restrict__ dst,
                                const float* __restrict__ es, const float* __restrict__ ed,
                                const float* __restrict__ emaxf, const float* __restrict__ denom,
                                const float* __restrict__ H2, float* __restrict__ MU) {
  int e = blockIdx.x * blockDim.x + threadIdx.x;
  if (e >= N_EDGES) return;
  const int s = src[e], d = dst[e];
  const float v = leaky_relu(es[s] + ed[d]);
  const float alpha = expf(v - emaxf[d]) / fmaxf(denom[d], 1e-16f);
  atomicAdd(&MU[d * 2 + 0], alpha * H2[s * 2 + 0]);
  atomicAdd(&MU[d * 2 + 1], alpha * H2[s * 2 + 1]);
}

// ---------- sampled-softmax logits: one block per node, 512 threads ----------
__global__ __launch_bounds__(512) void logits_kernel(const float* __restrict__ MU,
                                                     const float* __restrict__ ssw,
                                                     const float* __restrict__ ssb,
                                                     const int* __restrict__ y,
                                                     const int* __restrict__ sids,
                                                     float* __restrict__ out) {
  const int n = blockIdx.x;
  const float mu0 = MU[n * 2], mu1 = MU[n * 2 + 1];
  const int j = threadIdx.x;
  const int sid = sids[j];
  out[(size_t)n * (1 + N_SAMP) + 1 + j] =
      mu0 * ssw[sid * 2] + mu1 * ssw[sid * 2 + 1] + ssb[sid];
  if (j == 0) {
    const int ty = y[n];
    out[(size_t)n * (1 + N_SAMP)] =
        mu0 * ssw[ty * 2] + mu1 * ssw[ty * 2 + 1] + ssb[ty];
  }
}

// ---------- host launcher ----------
extern "C" void kernel_launch(void* const* d_in, const int* in_sizes, int n_in,
                              void* d_out, int out_size, void* d_ws, size_t ws_size,
                              hipStream_t stream) {
  const float* X     = (const float*)d_in[0];
  const int*   y     = (const int*)d_in[2];
  const int*   eidx  = (const int*)d_in[3];
  const int*   sids  = (const int*)d_in[4];
  const float* W1    = (const float*)d_in[5];
  const float* a1s   = (const float*)d_in[6];
  const float* a1d   = (const float*)d_in[7];
  const float* Wmu   = (const float*)d_in[8];
  const float* amus  = (const float*)d_in[9];
  const float* amud  = (const float*)d_in[10];
  const float* ssw   = (const float*)d_in[14];
  const float* ssb   = (const float*)d_in[15];

  const int* src = eidx;
  const int* dst = eidx + N_EDGES;

  // workspace layout (bytes)
  char* ws = (char*)d_ws;
  float*    H     = (float*)(ws + 0);                       // N*128 f32 = 51.2 MB
  float*    AGG   = (float*)(ws + 51200000);                // N*128 f32 = 51.2 MB
  float*    H2    = (float*)(ws + 102400000);               // N*2 f32
  float*    ES    = (float*)(ws + 103200000);               // N f32 (reused layer 2)
  float*    ED    = (float*)(ws + 103600000);               // N f32 (reused layer 2)
  unsigned* EMAX  = (unsigned*)(ws + 104000000);            // N u32 -> f32 in place
  float*    DEN   = (float*)(ws + 104400000);               // N f32
  float*    logits = (float*)d_out;                         // N*513
  float*    MU     = (float*)d_out + (size_t)N_NODES * (1 + N_SAMP);  // N*2 tail

  const int T = 256;
  const int gN  = (N_NODES + T - 1) / T;
  const int gE  = (N_EDGES + T - 1) / T;
  const int gNH = (N_NODES * F_HID + T - 1) / T;
  const int gN8 = N_NODES / 8;      // 12500
  const int gE8 = N_EDGES / 8;      // 200000

  // ---- init (harness poisons buffers; re-init every call) ----
  fill_f32<<<gNH, T, 0, stream>>>(AGG, 0.f, N_NODES * F_HID);
  fill_f32<<<gN,  T, 0, stream>>>(DEN, 0.f, N_NODES);
  fill_u32<<<gN,  T, 0, stream>>>(EMAX, ORD_NEG_INF, N_NODES);
  fill_f32<<<(N_NODES * 2 + T - 1) / T, T, 0, stream>>>(MU, 0.f, N_NODES * 2);

  // ---- layer 1 ----
  gemm_x_w1<<<N_NODES / 16, 256, 0, stream>>>(X, W1, H);
  node_edge_feats<<<gN8, 256, 0, stream>>>(H, a1s, a1d, ES, ED);
  edge_max<<<gE, T, 0, stream>>>(src, dst, ES, ED, EMAX);
  finalize_max<<<gN, T, 0, stream>>>(EMAX, N_NODES);
  edge_expsum<<<gE, T, 0, stream>>>(src, dst, ES, ED, (float*)EMAX, DEN);
  edge_aggregate128<<<gE8, 256, 0, stream>>>(src, dst, ES, ED, (float*)EMAX, DEN, H, AGG);

  // ---- layer 2 (mu) ----
  node_h2<<<gN8, 256, 0, stream>>>(AGG, Wmu, amus, amud, H2, ES, ED);  // ES/ED reused
  fill_f32<<<gN, T, 0, stream>>>(DEN, 0.f, N_NODES);
  fill_u32<<<gN, T, 0, stream>>>(EMAX, ORD_NEG_INF, N_NODES);
  edge_max<<<gE, T, 0, stream>>>(src, dst, ES, ED, EMAX);
  finalize_max<<<gN, T, 0, stream>>>(EMAX, N_NODES);
  edge_expsum<<<gE, T, 0, stream>>>(src, dst, ES, ED, (float*)EMAX, DEN);
  edge_aggregate2<<<gE, T, 0, stream>>>(src, dst, ES, ED, (float*)EMAX, DEN, H2, MU);

  // ---- sampled softmax logits ----
  logits_kernel<<<N_NODES, 512, 0, stream>>>(MU, ssw, ssb, y, sids, logits);

  (void)in_sizes; (void)n_in; (void)out_size; (void)ws_size;
}